// DebugMultiHeadAttention_44332652429420
// MI455X (gfx1250) — compile-verified
//
#include <hip/hip_runtime.h>
#include <hip/hip_bf16.h>

typedef __attribute__((ext_vector_type(2))) float v2f;
typedef __attribute__((ext_vector_type(8))) float v8f;

constexpr int kN    = 4096;     // nodes
constexpr int kE    = 131072;   // edges
constexpr int kDin  = 1024;
constexpr int kD    = 1024;
constexpr int kH    = 8;
constexpr int kDh   = 128;
constexpr int kNT   = kN / 16;  // 256 row/col tiles
constexpr float kScale = 0.088388347648318447f; // 1/sqrt(128)

// ---------------------------------------------------------------------------
// f32 WMMA: D(16x16) = A(16x4) * B(4x16) + C.  A,B = 2 VGPRs; C/D = 8 VGPRs.
// Layout (ISA 7.12.2): A/B lane L<16 -> row/col L, vgpr0={K0}, vgpr1={K1};
// lanes 16..31 -> K+2.  C/D vgpr v: row v (lanes 0-15), row v+8 (lanes 16-31).
// ---------------------------------------------------------------------------
__device__ __forceinline__ v8f wmma4(v2f a, v2f b, v8f c) {
  return __builtin_amdgcn_wmma_f32_16x16x4_f32(
      /*neg_a=*/false, a, /*neg_b=*/false, b,
      /*c_mod=*/(short)0, c, /*reuse_a=*/false, /*reuse_b=*/false);
}

__device__ __forceinline__ float softplus_f(float x) {
  return (x > 20.f) ? x : log1pf(expf(x));
}

// ------------------------------ mask build ---------------------------------
__global__ void mask_init(float4* __restrict__ m, int n4) {
  int i = blockIdx.x * blockDim.x + threadIdx.x;
  int stride = gridDim.x * blockDim.x;
  const float ninf = -__builtin_inff();
  float4 v = make_float4(ninf, ninf, ninf, ninf);
  for (; i < n4; i += stride) m[i] = v;
}

__global__ void mask_scatter(const int* __restrict__ ei, const int* __restrict__ et,
                             const float* __restrict__ etw, float* __restrict__ mask) {
  int e = blockIdx.x * blockDim.x + threadIdx.x;
  if (e >= kE) return;
  int src = ei[e];
  int dst = ei[kE + e];
  int t = et[e] - 1;
  mask[(size_t)src * kN + dst] = softplus_f(etw[t]);
}

__global__ void mask_diag(const float* __restrict__ etw, float* __restrict__ mask) {
  int i = blockIdx.x * blockDim.x + threadIdx.x;
  if (i >= kN) return;
  size_t p = (size_t)i * kN + i;
  if (mask[p] == -__builtin_inff()) mask[p] = softplus_f(etw[3]);
}

// ------------------------------ GEMM ---------------------------------------
// C[M x Nc] = A[M x K] * op(B) + bias.
// OPB==1: B stored [Nc x K] row-major (C = A * B^T)   -> x @ W_eff^T
// OPB==0: B stored [K x Nc] row-major (C = A * B)     -> w2 @ W1
// One 16x64 C strip per wave (4 independent accumulator tiles).  K consumed in
// chunks of 32: fragments staged into register arrays so the ~40 loads issue
// as clauses, then 32 interleaved v_wmma_f32_16x16x4_f32.
template <int OPB>
__global__ __launch_bounds__(256) void gemm_wmma(const float* __restrict__ A,
                                                 const float* __restrict__ B,
                                                 const float* __restrict__ bias,
                                                 float* __restrict__ C,
                                                 int M, int Ncols, int K) {
  const int wave = threadIdx.x >> 5;
  const int lane = threadIdx.x & 31;
  const int tile = blockIdx.x * 8 + wave;
  const int tilesN = Ncols >> 6;          // 64-column groups
  const int tm = tile / tilesN;
  const int tn = tile - tm * tilesN;
  if (tm * 16 >= M) return;
  const int r = lane & 15;
  const int hi = lane >> 4;

  const float* Arow = A + (size_t)(tm * 16 + r) * K + 2 * hi;
  v8f acc[4] = {};
  for (int k0 = 0; k0 < K; k0 += 32) {   // 8 k-steps per chunk
    v2f a8[8];
#pragma unroll
    for (int s = 0; s < 8; ++s) {
      a8[s].x = Arow[k0 + 4 * s];
      a8[s].y = Arow[k0 + 4 * s + 1];
    }
    v2f b8[4][8];
    if (OPB) {
      const float* B0 = B + (size_t)(tn * 64 + r) * K + 2 * hi + k0;
#pragma unroll
      for (int j = 0; j < 4; ++j)
#pragma unroll
        for (int s = 0; s < 8; ++s) {
          const float* Bp = B0 + (size_t)(j * 16) * K + 4 * s;
          b8[j][s].x = Bp[0];
          b8[j][s].y = Bp[1];
        }
    } else {
      const float* B0 = B + (size_t)(k0 + 2 * hi) * Ncols + tn * 64 + r;
#pragma unroll
      for (int j = 0; j < 4; ++j)
#pragma unroll
        for (int s = 0; s < 8; ++s) {
          const float* Bp = B0 + (size_t)(4 * s) * Ncols + j * 16;
          b8[j][s].x = Bp[0];
          b8[j][s].y = Bp[Ncols];
        }
    }
#pragma unroll
    for (int s = 0; s < 8; ++s)
#pragma unroll
      for (int j = 0; j < 4; ++j)
        acc[j] = wmma4(a8[s], b8[j][s], acc[j]);
  }
#pragma unroll
  for (int j = 0; j < 4; ++j) {
    float bv = (bias != nullptr) ? bias[tn * 64 + j * 16 + r] : 0.f;
    float* Cp = C + (size_t)(tm * 16 + 8 * hi) * Ncols + tn * 64 + j * 16 + r;
#pragma unroll
    for (int v = 0; v < 8; ++v) Cp[(size_t)v * Ncols] = acc[j][v] + bv;
  }
}

// b_eff[m] = sum_j w2[m,j]*b1[j] + b2[m]
__global__ void compose_bias(const float* __restrict__ w2, const float* __restrict__ b1,
                             const float* __restrict__ b2, float* __restrict__ beff) {
  int m = blockIdx.x * blockDim.x + threadIdx.x;
  if (m >= kD) return;
  float s = b2[m];
  const float* row = w2 + (size_t)m * kD;
  for (int j = 0; j < kD; ++j) s += row[j] * b1[j];
  beff[m] = s;
}

// --------------------- attention pass 1: row max / sumexp ------------------
// grid: (H * 256)/8 blocks of 8 waves; each wave owns one (head, 16-row tile)
// and streams over all 256 key tiles with online-softmax accumulation.
// K fragments are staged in half-tile register arrays and software-pipelined
// across tiles: next half-tile loads (and mask fragments) are in flight while
// WMMA + the shuffle/exp reduction run.
__global__ __launch_bounds__(256) void attn_pass1(const float* __restrict__ Qh,
                                                  const float* __restrict__ Kh,
                                                  const float* __restrict__ mask,
                                                  float* __restrict__ rowmax,
                                                  float* __restrict__ rowsum) {
  const int wave = threadIdx.x >> 5;
  const int lane = threadIdx.x & 31;
  const int gw = blockIdx.x * 8 + wave;
  const int h = gw >> 8;     // / 256
  const int nt = gw & 255;
  const int r = lane & 15;
  const int hi = lane >> 4;
  const float ninf = -__builtin_inff();

  const float* Qrow = Qh + (size_t)(nt * 16 + r) * kD + h * kDh;
  v2f aq[32];
#pragma unroll
  for (int kk = 0; kk < 32; ++kk) {
    const float* p = Qrow + kk * 4 + 2 * hi;
    aq[kk].x = p[0] * kScale;
    aq[kk].y = p[1] * kScale;
  }

  const float* Kbase = Kh + (size_t)r * kD + h * kDh + 2 * hi;
  auto loadHalf = [&](int mt, int half, v2f(&bk)[16]) {
    const float* Krow = Kbase + (size_t)(mt * 16) * kD + half * 64;
#pragma unroll
    for (int kk = 0; kk < 16; ++kk) {
      bk[kk].x = Krow[kk * 4];
      bk[kk].y = Krow[kk * 4 + 1];
    }
  };

  float m_run[8], l_run[8];
#pragma unroll
  for (int v = 0; v < 8; ++v) { m_run[v] = ninf; l_run[v] = 0.f; }

  v2f b0[16], b1[16];
  loadHalf(0, 0, b0);
  for (int mt = 0; mt < kNT; ++mt) {
    loadHalf(mt, 1, b1);
    const float* mrow = mask + (size_t)(nt * 16 + 8 * hi) * kN + mt * 16 + r;
    float mfrag[8];
#pragma unroll
    for (int v = 0; v < 8; ++v) mfrag[v] = mrow[(size_t)v * kN];

    v8f acc0 = {}, acc1 = {};
#pragma unroll
    for (int kk = 0; kk < 16; kk += 2) {
      acc0 = wmma4(aq[kk], b0[kk], acc0);
      acc1 = wmma4(aq[kk + 1], b0[kk + 1], acc1);
    }
    if (mt + 1 < kNT) loadHalf(mt + 1, 0, b0);   // prefetch next tile
#pragma unroll
    for (int kk = 0; kk < 16; kk += 2) {
      acc0 = wmma4(aq[16 + kk], b1[kk], acc0);
      acc1 = wmma4(aq[16 + kk + 1], b1[kk + 1], acc1);
    }
#pragma unroll
    for (int v = 0; v < 8; ++v) {
      float s = acc0[v] + acc1[v] + mfrag[v];
      float tmax = s;
#pragma unroll
      for (int off = 1; off < 16; off <<= 1)
        tmax = fmaxf(tmax, __shfl_xor(tmax, off, 32));
      float e = (tmax > ninf) ? __expf(s - tmax) : 0.f;
      float tsum = e;
#pragma unroll
      for (int off = 1; off < 16; off <<= 1)
        tsum += __shfl_xor(tsum, off, 32);
      if (tmax > ninf) {
        float nm = fmaxf(m_run[v], tmax);
        l_run[v] = l_run[v] * __expf(m_run[v] - nm) + tsum * __expf(tmax - nm);
        m_run[v] = nm;
      }
    }
  }
  if (r == 0) {
#pragma unroll
    for (int v = 0; v < 8; ++v) {
      int row = nt * 16 + 8 * hi + v;
      rowmax[h * kN + row] = m_run[v];
      rowsum[h * kN + row] = l_run[v];
    }
  }
}

// --------------------- attention pass 2: ctx + head-mean -------------------
// grid: 256 blocks (one per 16-row query tile), 8 waves = 8 heads.
// Per key tile: recompute scores (WMMA), p = exp(s-m)/l; accumulate p/8 into a
// shared tile via ds_add_f32 atomics (head-mean output) and ctx via WMMA with
// the P tile transposed into A-operand layout through wave-private LDS.
__global__ __launch_bounds__(256) void attn_pass2(const float* __restrict__ Qh,
                                                  const float* __restrict__ Kh,
                                                  const float* __restrict__ Vh,
                                                  const float* __restrict__ mask,
                                                  const float* __restrict__ rowmax,
                                                  const float* __restrict__ rowsum,
                                                  float* __restrict__ ctx,
                                                  float* __restrict__ attn_w) {
  __shared__ float pT[kH][16][17];
  __shared__ float accW[16][17];
  const int h = threadIdx.x >> 5;  // wave == head
  const int lane = threadIdx.x & 31;
  const int nt = blockIdx.x;
  const int r = lane & 15;
  const int hi = lane >> 4;
  const int tr = threadIdx.x >> 4;  // 0..15 (tile store helpers)
  const int tc = threadIdx.x & 15;

  const float* Qrow = Qh + (size_t)(nt * 16 + r) * kD + h * kDh;
  v2f aq[32];
#pragma unroll
  for (int kk = 0; kk < 32; ++kk) {
    const float* p = Qrow + kk * 4 + 2 * hi;
    aq[kk].x = p[0] * kScale;
    aq[kk].y = p[1] * kScale;
  }
  float m8[8], li8[8];
#pragma unroll
  for (int v = 0; v < 8; ++v) {
    int row = nt * 16 + 8 * hi + v;
    m8[v] = rowmax[h * kN + row];
    float l = rowsum[h * kN + row];
    li8[v] = (l > 0.f) ? 1.f / l : 0.f;
  }

  const float* Kbase = Kh + (size_t)r * kD + h * kDh + 2 * hi;
  auto loadHalf = [&](int mt, int half, v2f(&bk)[16]) {
    const float* Krow = Kbase + (size_t)(mt * 16) * kD + half * 64;
#pragma unroll
    for (int kk = 0; kk < 16; ++kk) {
      bk[kk].x = Krow[kk * 4];
      bk[kk].y = Krow[kk * 4 + 1];
    }
  };

  v8f cacc[8] = {};  // ctx accumulators: 8 d-tiles of 16 (DH=128)

  for (int mt = 0; mt < kNT; ++mt) {
    __syncthreads();          // prior iter's pT reads + accW stores done
    accW[tr][tc] = 0.f;

    v2f b0[16], b1[16];
    loadHalf(mt, 0, b0);
    loadHalf(mt, 1, b1);
    const float* mrow = mask + (size_t)(nt * 16 + 8 * hi) * kN + mt * 16 + r;
    float mfrag[8];
#pragma unroll
    for (int v = 0; v < 8; ++v) mfrag[v] = mrow[(size_t)v * kN];

    v8f acc0 = {}, acc1 = {};
#pragma unroll
    for (int kk = 0; kk < 16; kk += 2) {
      acc0 = wmma4(aq[kk], b0[kk], acc0);
      acc1 = wmma4(aq[kk + 1], b0[kk + 1], acc1);
    }
#pragma unroll
    for (int kk = 0; kk < 16; kk += 2) {
      acc0 = wmma4(aq[16 + kk], b1[kk], acc0);
      acc1 = wmma4(aq[16 + kk + 1], b1[kk + 1], acc1);
    }
#pragma unroll
    for (int v = 0; v < 8; ++v) {
      float s = acc0[v] + acc1[v] + mfrag[v];
      float pv = __expf(s - m8[v]) * li8[v];
      pT[h][8 * hi + v][r] = pv;
      atomicAdd(&accW[8 * hi + v][r], pv * 0.125f);  // mean over 8 heads
    }
    __syncthreads();          // pT + accW complete

    // ctx += P(16x16) * V(16x128): P re-read from LDS in A-operand layout
    v2f pa[4];
#pragma unroll
    for (int kk = 0; kk < 4; ++kk) {
      pa[kk].x = pT[h][r][4 * kk + 2 * hi];
      pa[kk].y = pT[h][r][4 * kk + 2 * hi + 1];
    }
    const float* Vbase = Vh + (size_t)(mt * 16) * kD + h * kDh + r;
#pragma unroll
    for (int kk = 0; kk < 4; ++kk) {
      const float* vr = Vbase + (size_t)(4 * kk + 2 * hi) * kD;
      v2f vb[8];
#pragma unroll
      for (int dt = 0; dt < 8; ++dt) {
        vb[dt].x = vr[dt * 16];
        vb[dt].y = vr[kD + dt * 16];
      }
#pragma unroll
      for (int dt = 0; dt < 8; ++dt)
        cacc[dt] = wmma4(pa[kk], vb[dt], cacc[dt]);
    }
    attn_w[(size_t)(nt * 16 + tr) * kN + mt * 16 + tc] = accW[tr][tc];
  }

#pragma unroll
  for (int dt = 0; dt < 8; ++dt)
#pragma unroll
    for (int v = 0; v < 8; ++v)
      ctx[(size_t)(nt * 16 + 8 * hi + v) * kD + h * kDh + dt * 16 + r] = cacc[dt][v];
}

// ---------------------------------------------------------------------------
extern "C" void kernel_launch(void* const* d_in, const int* in_sizes, int n_in,
                              void* d_out, int out_size, void* d_ws, size_t ws_size,
                              hipStream_t stream) {
  (void)in_sizes; (void)n_in; (void)out_size; (void)ws_size;
  const float* x   = (const float*)d_in[0];
  const int*   ei  = (const int*)d_in[1];
  const int*   et  = (const int*)d_in[2];
  const float* etw = (const float*)d_in[3];
  const float* W1[3] = {(const float*)d_in[4], (const float*)d_in[6], (const float*)d_in[8]};
  const float* b1[3] = {(const float*)d_in[5], (const float*)d_in[7], (const float*)d_in[9]};
  const float* in_w  = (const float*)d_in[10];
  const float* in_b  = (const float*)d_in[11];
  const float* out_w = (const float*)d_in[12];
  const float* out_b = (const float*)d_in[13];

  float* out       = (float*)d_out;                  // N*D
  float* attn_w    = out + (size_t)kN * kD;          // N*N
  float* attn_mask = attn_w + (size_t)kN * kN;       // N*N

  float* Weff = (float*)d_ws;                        // 3 * D*DIN
  float* beff = Weff + 3 * (size_t)kD * kDin;        // 3 * D
  float* Qh   = beff + 3 * kD;                       // N*D each
  float* Kh   = Qh + (size_t)kN * kD;
  float* Vh   = Kh + (size_t)kN * kD;
  float* ctx  = Vh + (size_t)kN * kD;
  float* rmax = ctx + (size_t)kN * kD;               // H*N
  float* rsum = rmax + (size_t)kH * kN;              // H*N

  // mask: init(-inf) -> edge scatter -> diagonal self-loop fix
  mask_init<<<2048, 256, 0, stream>>>((float4*)attn_mask, kN * kN / 4);
  mask_scatter<<<kE / 256, 256, 0, stream>>>(ei, et, etw, attn_mask);
  mask_diag<<<kN / 256, 256, 0, stream>>>(etw, attn_mask);

  // fused projections: W_eff = w2 @ W1 ; b_eff = w2 @ b1 + b2 ; Xh = x @ W_eff^T + b_eff
  float* QKVh[3] = {Qh, Kh, Vh};
  for (int i = 0; i < 3; ++i) {
    gemm_wmma<0><<<(64 * 16) / 8, 256, 0, stream>>>(
        in_w + (size_t)i * kD * kD, W1[i], nullptr,
        Weff + (size_t)i * kD * kDin, kD, kDin, kD);
    compose_bias<<<kD / 256, 256, 0, stream>>>(
        in_w + (size_t)i * kD * kD, b1[i], in_b + i * kD, beff + i * kD);
    gemm_wmma<1><<<(kNT * 16) / 8, 256, 0, stream>>>(
        x, Weff + (size_t)i * kD * kDin, beff + i * kD, QKVh[i], kN, kD, kDin);
  }

  attn_pass1<<<(kH * kNT) / 8, 256, 0, stream>>>(Qh, Kh, attn_mask, rmax, rsum);
  attn_pass2<<<kNT, 256, 0, stream>>>(Qh, Kh, Vh, attn_mask, rmax, rsum, ctx, attn_w);

  // out = ctx @ out_w^T + out_b
  gemm_wmma<1><<<(kNT * 16) / 8, 256, 0, stream>>>(ctx, out_w, out_b, out, kN, kD, kD);
}